// GDN_20306605375641
// MI455X (gfx1250) — compile-verified
//
#include <hip/hip_runtime.h>
#include <math.h>

// ---- problem constants (from reference setup_inputs) ----
#define Bsz  32
#define Nn   512
#define Ff   64
#define Dd   64
#define Kk   20
#define BN   (Bsz * Nn)        // 16384 nodes
#define BNEPS 1e-5f
#define NSLOPE 0.2f

typedef float v2f __attribute__((ext_vector_type(2)));
typedef float v8f __attribute__((ext_vector_type(8)));

// -------------------------------------------------------------------------
// K0: inverse norms of embedding rows: invn[i] = 1/||emb_i||
// -------------------------------------------------------------------------
__global__ void gdn_norm_kernel(const float* __restrict__ emb,
                                float* __restrict__ invn) {
    const int i = blockIdx.x * blockDim.x + threadIdx.x;
    if (i >= Nn) return;
    float ssq = 0.f;
    #pragma unroll
    for (int d = 0; d < Dd; ++d) { float v = emb[i * Dd + d]; ssq += v * v; }
    invn[i] = 1.0f / sqrtf(ssq);
}

// -------------------------------------------------------------------------
// K1a: cos = (emb @ emb^T) normalized, via V_WMMA_F32_16X16X4_F32.
// C[i][j] = dot(emb_i, emb_j) * invn[i] * invn[j].
// A fragment: lane l -> M = l&15, VGPR0 K=2*(l>>4), VGPR1 K=2*(l>>4)+1.
// B = emb^T: B[k][j] = emb[j*D+k] -> same load form as A with j on lanes.
// One wave per 16x16 tile, grid (N/16, N/16).
// -------------------------------------------------------------------------
__global__ void gdn_cos_wmma_kernel(const float* __restrict__ emb,
                                    const float* __restrict__ invn,
                                    float* __restrict__ cosm) {
    const int iBase = blockIdx.x * 16;
    const int jBase = blockIdx.y * 16;
    const int lane  = threadIdx.x;
    const int half  = lane >> 4;
    const int l16   = lane & 15;

    const float* ar = emb + (iBase + l16) * Dd + 2 * half;
    const float* br = emb + (jBase + l16) * Dd + 2 * half;

    v8f c = {};
    #pragma unroll
    for (int k0 = 0; k0 < Dd; k0 += 4) {
        v2f a, b;
        a.x = ar[k0]; a.y = ar[k0 + 1];
        b.x = br[k0]; b.y = br[k0 + 1];
        c = __builtin_amdgcn_wmma_f32_16x16x4_f32(
                false, a, false, b, (short)0, c, false, false);
    }
    const int col = jBase + l16;
    const float inv_j = invn[col];
    #pragma unroll
    for (int r = 0; r < 8; ++r) {
        const int row = iBase + r + 8 * half;
        cosm[row * Nn + col] = c[r] * invn[row] * inv_j;
    }
}

// -------------------------------------------------------------------------
// K1b: per-row top-20 selection (tie-break: lowest index, matching
// jax.lax.top_k). One 256-thread block per row; cos row staged in LDS.
// -------------------------------------------------------------------------
__global__ void gdn_topk_kernel(const float* __restrict__ cosm,
                                int* __restrict__ topk) {
    const int i = blockIdx.x;
    const int tid = threadIdx.x;
    __shared__ float cosv[Nn];
    __shared__ float redv[256];
    __shared__ int   redi[256];

    for (int j = tid; j < Nn; j += 256) cosv[j] = cosm[i * Nn + j];
    __syncthreads();

    for (int k = 0; k < Kk; ++k) {
        float bv = -3.0e38f; int bi = Nn;
        for (int j = tid; j < Nn; j += 256) {
            float v = cosv[j];
            if (v > bv || (v == bv && j < bi)) { bv = v; bi = j; }
        }
        redv[tid] = bv; redi[tid] = bi;
        __syncthreads();
        for (int s = 128; s > 0; s >>= 1) {
            if (tid < s) {
                float ov = redv[tid + s]; int oi = redi[tid + s];
                if (ov > redv[tid] || (ov == redv[tid] && oi < redi[tid])) {
                    redv[tid] = ov; redi[tid] = oi;
                }
            }
            __syncthreads();
        }
        if (tid == 0) {
            topk[i * Kk + k] = redi[0];
            cosv[redi[0]] = -3.0e38f;   // exclude from next round
        }
        __syncthreads();
    }
}

// -------------------------------------------------------------------------
// K2: xl = x @ lin_W^T via V_WMMA_F32_16X16X4_F32.
// x: [BN, F] row-major. lin_W: [D, F] row-major -> B[k][n] = lin_W[n*F+k].
// One wave per 16x16 output tile; grid (BN/16, D/16).
// -------------------------------------------------------------------------
__global__ void gdn_xl_wmma_kernel(const float* __restrict__ x,
                                   const float* __restrict__ W,
                                   float* __restrict__ xl) {
    const int mBase = blockIdx.x * 16;
    const int nBase = blockIdx.y * 16;
    const int lane  = threadIdx.x;
    const int half  = lane >> 4;
    const int l16   = lane & 15;

    const float* xr = x + (mBase + l16) * Ff + 2 * half;
    const float* wr = W + (nBase + l16) * Ff + 2 * half;

    v8f c = {};
    #pragma unroll
    for (int k0 = 0; k0 < Ff; k0 += 4) {
        v2f a, b;
        a.x = xr[k0]; a.y = xr[k0 + 1];
        b.x = wr[k0]; b.y = wr[k0 + 1];
        c = __builtin_amdgcn_wmma_f32_16x16x4_f32(
                false, a, false, b, (short)0, c, false, false);
    }
    #pragma unroll
    for (int r = 0; r < 8; ++r) {
        xl[(mBase + r + 8 * half) * Dd + nBase + l16] = c[r];
    }
}

// -------------------------------------------------------------------------
// K3: per-node attention scalars:
//   s_i[n] = xl[n]·att_i + emb[n%N]·att_em_i
//   s_j[n] = xl[n]·att_j + emb[n%N]·att_em_j
// -------------------------------------------------------------------------
__global__ void gdn_score_kernel(const float* __restrict__ xl,
                                 const float* __restrict__ emb,
                                 const float* __restrict__ att_i,
                                 const float* __restrict__ att_j,
                                 const float* __restrict__ att_em_i,
                                 const float* __restrict__ att_em_j,
                                 float* __restrict__ s_i,
                                 float* __restrict__ s_j) {
    const int n = blockIdx.x * blockDim.x + threadIdx.x;
    if (n >= BN) return;
    const int node = n & (Nn - 1);
    float si = 0.f, sj = 0.f;
    #pragma unroll
    for (int d = 0; d < Dd; ++d) {
        float xv = xl[n * Dd + d];
        float ev = emb[node * Dd + d];
        si += xv * att_i[d] + ev * att_em_i[d];
        sj += xv * att_j[d] + ev * att_em_j[d];
    }
    s_i[n] = si;
    s_j[n] = sj;
}

// -------------------------------------------------------------------------
// K4: fused per-target-node pipeline. One 64-thread block per node (tid = d):
//   alpha_k = leakyrelu(s_i[dst] + s_j[src_k]); softmax over k=0..19;
//   agg[d] = sum_k a_k * xl[src_k][d] + gnn_bias[d];
//   h = relu(BN1(agg)); out_features = h * emb[i][d];
//   o = relu(BN2(out_features)); out[node] = o·outW + outb
// -------------------------------------------------------------------------
__global__ void gdn_node_kernel(const float* __restrict__ xl,
                                const float* __restrict__ emb,
                                const int*   __restrict__ topk,
                                const float* __restrict__ s_i,
                                const float* __restrict__ s_j,
                                const float* __restrict__ gnn_bias,
                                const float* __restrict__ bn1g, const float* __restrict__ bn1b,
                                const float* __restrict__ bn1m, const float* __restrict__ bn1v,
                                const float* __restrict__ bn2g, const float* __restrict__ bn2b,
                                const float* __restrict__ bn2m, const float* __restrict__ bn2v,
                                const float* __restrict__ outW, const float* __restrict__ outb,
                                float* __restrict__ out_features,
                                float* __restrict__ outvec) {
    const int nodeG = blockIdx.x;          // 0..BN-1
    const int b     = nodeG >> 9;          // / Nn
    const int i     = nodeG & (Nn - 1);
    const int tid   = threadIdx.x;         // 0..63 == d

    __shared__ int   srcS[Kk];
    __shared__ float aw[Kk];
    __shared__ float red[Dd];

    if (tid < Kk) {
        int s = topk[i * Kk + tid] + b * Nn;
        srcS[tid] = s;
        float al = s_i[nodeG] + s_j[s];
        al = (al > 0.f) ? al : NSLOPE * al;     // leaky relu
        aw[tid] = al;
    }
    __syncthreads();
    if (tid == 0) {
        float m = aw[0];
        #pragma unroll
        for (int k = 1; k < Kk; ++k) m = fmaxf(m, aw[k]);
        float sum = 0.f;
        #pragma unroll
        for (int k = 0; k < Kk; ++k) { float e = expf(aw[k] - m); aw[k] = e; sum += e; }
        float inv = 1.0f / sum;
        #pragma unroll
        for (int k = 0; k < Kk; ++k) aw[k] *= inv;
    }
    __syncthreads();

    const int d = tid;
    float acc = 0.f;
    #pragma unroll
    for (int k = 0; k < Kk; ++k) acc += aw[k] * xl[srcS[k] * Dd + d];
    acc += gnn_bias[d];

    float h = (acc - bn1m[d]) * rsqrtf(bn1v[d] + BNEPS) * bn1g[d] + bn1b[d];
    h = fmaxf(h, 0.f);
    float of = h * emb[i * Dd + d];
    out_features[nodeG * Dd + d] = of;

    float o = (of - bn2m[d]) * rsqrtf(bn2v[d] + BNEPS) * bn2g[d] + bn2b[d];
    o = fmaxf(o, 0.f);
    red[tid] = o * outW[d];
    __syncthreads();
    for (int s2 = 32; s2 > 0; s2 >>= 1) {
        if (tid < s2) red[tid] += red[tid + s2];
        __syncthreads();
    }
    if (tid == 0) outvec[nodeG] = red[0] + outb[0];
}

// -------------------------------------------------------------------------
// K5: deterministic single-block MSE reduction: loss = mean((out-labels)^2)
// -------------------------------------------------------------------------
__global__ void gdn_loss_kernel(const float* __restrict__ outvec,
                                const float* __restrict__ labels,
                                float* __restrict__ loss) {
    __shared__ float red[256];
    const int tid = threadIdx.x;
    float acc = 0.f;
    for (int j = tid; j < BN; j += 256) {
        float dlt = outvec[j] - labels[j];
        acc += dlt * dlt;
    }
    red[tid] = acc;
    __syncthreads();
    for (int s = 128; s > 0; s >>= 1) {
        if (tid < s) red[tid] += red[tid + s];
        __syncthreads();
    }
    if (tid == 0) loss[0] = red[0] / (float)BN;
}

// -------------------------------------------------------------------------
extern "C" void kernel_launch(void* const* d_in, const int* in_sizes, int n_in,
                              void* d_out, int out_size, void* d_ws, size_t ws_size,
                              hipStream_t stream) {
    const float* data     = (const float*)d_in[0];   // [B,N,F]
    // d_in[1] = org_edge_index (unused)
    const float* labels   = (const float*)d_in[2];   // [B,N]
    const float* emb      = (const float*)d_in[3];   // [N,D]
    const float* lin_W    = (const float*)d_in[4];   // [D,F]
    const float* att_i    = (const float*)d_in[5];
    const float* att_j    = (const float*)d_in[6];
    const float* att_em_i = (const float*)d_in[7];
    const float* att_em_j = (const float*)d_in[8];
    const float* gnn_bias = (const float*)d_in[9];
    const float* bn1g = (const float*)d_in[10];
    const float* bn1b = (const float*)d_in[11];
    const float* bn1m = (const float*)d_in[12];
    const float* bn1v = (const float*)d_in[13];
    const float* bn2g = (const float*)d_in[14];
    const float* bn2b = (const float*)d_in[15];
    const float* bn2m = (const float*)d_in[16];
    const float* bn2v = (const float*)d_in[17];
    const float* outW = (const float*)d_in[18];      // [1,D]
    const float* outb = (const float*)d_in[19];      // [1]

    // output layout: [out_features (BN*D) | out (BN) | loss (1)]
    float* out_features = (float*)d_out;
    float* outvec       = out_features + (size_t)BN * Dd;
    float* loss         = outvec + BN;

    // workspace layout
    char* ws = (char*)d_ws;
    int*   topk = (int*)ws;                                  // N*K ints (40960 B)
    float* invn = (float*)(ws + 48 * 1024);                  // N floats
    float* cosm = (float*)(ws + 64 * 1024);                  // N*N floats (1 MB)
    float* xl   = (float*)(ws + 2 * 1024 * 1024);            // BN*D floats (4 MB)
    float* s_i  = (float*)(ws + 6 * 1024 * 1024);            // BN floats
    float* s_j  = s_i + BN;                                  // BN floats

    // K0: inverse embedding norms
    gdn_norm_kernel<<<(Nn + 255) / 256, 256, 0, stream>>>(emb, invn);

    // K1a: cosine similarity matrix via WMMA (normalization fused in store)
    dim3 gc(Nn / 16, Nn / 16);
    gdn_cos_wmma_kernel<<<gc, 32, 0, stream>>>(emb, invn, cosm);

    // K1b: learned top-k graph (selection only)
    gdn_topk_kernel<<<Nn, 256, 0, stream>>>(cosm, topk);

    // K2: xl = x @ lin_W^T (WMMA f32 16x16x4)
    dim3 g2(BN / 16, Dd / 16);
    gdn_xl_wmma_kernel<<<g2, 32, 0, stream>>>(data, lin_W, xl);

    // K3: per-node attention scalars
    gdn_score_kernel<<<BN / 256, 256, 0, stream>>>(xl, emb, att_i, att_j,
                                                   att_em_i, att_em_j, s_i, s_j);

    // K4: fused softmax-aggregate + BN1/ReLU + emb-mul + BN2/ReLU + OutLayer
    gdn_node_kernel<<<BN, Dd, 0, stream>>>(xl, emb, topk, s_i, s_j, gnn_bias,
                                           bn1g, bn1b, bn1m, bn1v,
                                           bn2g, bn2b, bn2m, bn2v,
                                           outW, outb, out_features, outvec);

    // K5: MSE loss (deterministic single-block reduction)
    gdn_loss_kernel<<<1, 256, 0, stream>>>(outvec, labels, loss);
}